// Concentration_4578435137606
// MI455X (gfx1250) — compile-verified
//
#include <hip/hip_runtime.h>
#include <math.h>

// Problem constants (from reference): B=64, A=32, E=256, H=128, K_FOCUS=12
#define BB 64
#define AA 32
#define EE 256
#define HH 128
#define KF 12
#define NPAIR (BB * AA)          // 2048
#define VE_STRIDE 129            // LDS row padding: (e*129+h)%64 == (e+h)%64 -> conflict-free

typedef __attribute__((ext_vector_type(2))) float v2f;
typedef __attribute__((ext_vector_type(8))) float v8f;

__device__ __forceinline__ v8f wmma_f32(v2f a, v2f b, v8f c) {
  // D = A(16x4,f32) * B(4x16,f32) + C(16x16,f32)
  return __builtin_amdgcn_wmma_f32_16x16x4_f32(
      /*neg_a=*/false, a, /*neg_b=*/false, b,
      /*c_mod=*/(short)0, c, /*reuse_a=*/false, /*reuse_b=*/false);
}

// CDNA5 async global->LDS copy (ASYNCcnt-tracked, no VGPR data staging).
// lds_addr: wave-relative LDS byte address; gaddr: 64-bit global address.
__device__ __forceinline__ void async_load_b128(unsigned int lds_addr,
                                                const float* gaddr) {
  asm volatile("global_load_async_to_lds_b128 %0, %1, off"
               :
               : "v"(lds_addr), "v"(gaddr)
               : "memory");
}

__device__ __forceinline__ void wait_asynccnt0() {
  asm volatile("s_wait_asynccnt 0x0" ::: "memory");
}

// C[M,128] = act(A[M,K] @ B + bias)
//   TRANSB=0: B row-major [K,128] (ldw = row length = 128)
//   TRANSB=1: B is W[N,K] row-major, used as W^T (ldw = K row length)
// grid.x = M/16 ; block = 256 (8 waves; wave w owns output cols [16w,16w+16))
template <int TRANSB, int RELU, int HASBIAS>
__global__ void gemm_wmma_kernel(const float* __restrict__ A, int lda,
                                 const float* __restrict__ W, int ldw,
                                 const float* __restrict__ bias,
                                 float* __restrict__ C, int ldc, int K) {
  const int lane = threadIdx.x & 31;
  const int wv   = threadIdx.x >> 5;   // 0..7 -> N tile
  const int half = lane >> 4;          // 0/1 (K sub-pair per ISA layout)
  const int r    = lane & 15;          // A row within tile / B column within tile
  const int m0   = blockIdx.x * 16;
  const int colg = wv * 16 + r;

  const float* arow = A + (size_t)(m0 + r) * lda + 2 * half;
  v8f acc = {0.f, 0.f, 0.f, 0.f, 0.f, 0.f, 0.f, 0.f};

  for (int kk = 0; kk < K; kk += 4) {
    // A fragment: lanes 0-15 hold (K=kk,kk+1), lanes 16-31 hold (kk+2,kk+3)
    v2f a = *(const v2f*)(arow + kk);
    v2f b;
    if (TRANSB) {
      const float* wp = W + (size_t)colg * ldw + kk + 2 * half;
      b.x = wp[0];
      b.y = wp[1];
    } else {
      const float* wp = W + (size_t)(kk + 2 * half) * ldw + colg;
      b.x = wp[0];
      b.y = wp[ldw];
    }
    acc = wmma_f32(a, b, acc);
  }

  float bv = HASBIAS ? bias[colg] : 0.0f;
#pragma unroll
  for (int j = 0; j < 8; ++j) {
    int row = m0 + j + 8 * half;   // C/D layout: VGPR j -> rows j and j+8
    float v = acc[j] + bv;
    if (RELU) v = fmaxf(v, 0.0f);
    C[(size_t)row * ldc + colg] = v;
  }
}

// One workgroup per (b,a) pair. Streams ve[b,a] (128 KB) into LDS via the
// CDNA5 async-to-LDS pipe, then: compat = norm * ve @ qk, mask, softmax,
// score@ve row-sum, top-12 gather.
__global__ void attn_kernel(const float* __restrict__ vs,
                            const float* __restrict__ ve,
                            const int* __restrict__ ve_dead,
                            const float* __restrict__ QK,
                            float* __restrict__ SV,    // [NPAIR,128]
                            float* __restrict__ VM,    // [NPAIR,256]
                            float* __restrict__ VC) {  // [NPAIR,1664]
  extern __shared__ float smem[];
  float* ve_s    = smem;                     // 256*129
  float* qk_s    = ve_s + EE * VE_STRIDE;    // 128
  float* score_s = qk_s + HH;                // 256
  float* redv    = score_s + EE;             // 256
  int*   redi    = (int*)(redv + EE);        // 256
  float* svp     = (float*)(redi + EE);      // 256
  int*   idx_sel = (int*)(svp + 256);        // 16

  const int tid = threadIdx.x;               // 256 threads
  const int pair = blockIdx.x;
  const float* vebase = ve + (size_t)pair * EE * HH;

  if (tid < HH) qk_s[tid] = QK[(size_t)pair * HH + tid];

  // Stream 128 KB tile into LDS: 32 async b128 transfers per thread
  // (ASYNCcnt max 32 outstanding per wave -- within the 6-bit counter).
  // Each lane targets its padded-row LDS slot directly; no ds_store traffic.
  for (int g = tid; g < EE * HH / 4; g += 256) {
    int e = g >> 5;              // 32 float4 per row
    int h = (g & 31) << 2;
    unsigned int lds_addr =
        (unsigned int)(size_t)(ve_s + e * VE_STRIDE + h);  // addr[31:0] = LDS byte addr
    async_load_b128(lds_addr, vebase + (size_t)g * 4);
  }
  wait_asynccnt0();
  __syncthreads();

  // compat_e (one entity per thread), mask dead -> -inf
  const float norm = 0.08838834764831845f;   // 1/sqrt(128)
  float c;
  {
    const float* row = ve_s + tid * VE_STRIDE;
    float acc = 0.f;
    for (int h = 0; h < HH; ++h) acc = fmaf(row[h], qk_s[h], acc);
    c = acc * norm;
    if (ve_dead[(size_t)pair * EE + tid] != 0) c = -__builtin_inff();
  }

  // softmax over E=256 (block reductions), nan_to_num for all-dead case
  redv[tid] = c;
  __syncthreads();
  for (int st = 128; st > 0; st >>= 1) {
    if (tid < st) redv[tid] = fmaxf(redv[tid], redv[tid + st]);
    __syncthreads();
  }
  float m = redv[0];
  __syncthreads();
  float p = (m == -__builtin_inff()) ? 0.f : __expf(c - m);
  redv[tid] = p;
  __syncthreads();
  for (int st = 128; st > 0; st >>= 1) {
    if (tid < st) redv[tid] += redv[tid + st];
    __syncthreads();
  }
  float s = redv[0];
  __syncthreads();
  float sc = (s > 0.f) ? p / s : 0.f;
  score_s[tid] = sc;
  __syncthreads();

  // sv[h] = sum_e score_e * ve[e,h]  (split E range across the two thread halves)
  {
    int h = tid & 127;
    int e0 = (tid >> 7) << 7;
    float acc = 0.f;
    for (int e = e0; e < e0 + 128; ++e)
      acc = fmaf(score_s[e], ve_s[e * VE_STRIDE + h], acc);
    svp[tid] = acc;
  }
  __syncthreads();
  if (tid < HH) SV[(size_t)pair * HH + tid] = svp[tid] + svp[tid + 128];

  // vs row -> first block of v_M and v_C stacks
  if (tid < HH) {
    float x = vs[(size_t)pair * HH + tid];
    VM[(size_t)pair * (2 * HH) + tid] = x;
    VC[(size_t)pair * ((KF + 1) * HH) + tid] = x;
  }

  // top-12 (iterative argmax, ties -> lower index, matching lax.top_k)
  for (int j = 0; j < KF; ++j) {
    redv[tid] = score_s[tid];
    redi[tid] = tid;
    __syncthreads();
    for (int st = 128; st > 0; st >>= 1) {
      if (tid < st) {
        float v2 = redv[tid + st];
        int i2 = redi[tid + st];
        if (v2 > redv[tid] || (v2 == redv[tid] && i2 < redi[tid])) {
          redv[tid] = v2;
          redi[tid] = i2;
        }
      }
      __syncthreads();
    }
    if (tid == 0) {
      int w = redi[0];
      idx_sel[j] = w;
      score_s[w] = -1.0f;   // scores are >= 0; mask out winner
    }
    __syncthreads();
  }

  // gather selected entity rows into VC[:, 128:1664]
  for (int t = tid; t < KF * HH; t += 256) {
    int j = t >> 7;
    int h = t & 127;
    VC[(size_t)pair * ((KF + 1) * HH) + (1 + j) * HH + h] =
        ve_s[idx_sel[j] * VE_STRIDE + h];
  }
}

static const size_t ATTN_LDS =
    (size_t)(EE * VE_STRIDE + HH + EE + EE + EE + 256 + 16) * sizeof(float); // ~136.8 KB

extern "C" void kernel_launch(void* const* d_in, const int* in_sizes, int n_in,
                              void* d_out, int out_size, void* d_ws, size_t ws_size,
                              hipStream_t stream) {
  (void)in_sizes; (void)n_in; (void)out_size; (void)ws_size;
  const float* vs      = (const float*)d_in[0];
  const float* ve      = (const float*)d_in[1];
  const int*   ve_dead = (const int*)d_in[2];
  const float* Wq      = (const float*)d_in[3];
  const float* Wk      = (const float*)d_in[4];
  const float* Wv      = (const float*)d_in[5];
  const float* W1      = (const float*)d_in[6];
  const float* b1      = (const float*)d_in[7];
  const float* W2      = (const float*)d_in[8];
  const float* b2      = (const float*)d_in[9];
  float* out = (float*)d_out;
  float* ws  = (float*)d_ws;

  // workspace layout (floats)
  float* Wqk = ws;                   // 128*128
  float* QK  = Wqk + HH * HH;        // 2048*128
  float* SV  = QK + NPAIR * HH;      // 2048*128
  float* VM  = SV + NPAIR * HH;      // 2048*256
  float* VC  = VM + NPAIR * 2 * HH;  // 2048*1664   (total ~17.9 MB)

  hipFuncSetAttribute(reinterpret_cast<const void*>(attn_kernel),
                      hipFuncAttributeMaxDynamicSharedMemorySize, (int)ATTN_LDS);

  // 1: Wqk = Wq @ Wk^T                       [128,128,128]
  gemm_wmma_kernel<1, 0, 0><<<dim3(HH / 16), 256, 0, stream>>>(
      Wq, HH, Wk, HH, nullptr, Wqk, HH, HH);
  // 1b: QK = vs @ Wqk                        [2048,128,128]
  gemm_wmma_kernel<0, 0, 0><<<dim3(NPAIR / 16), 256, 0, stream>>>(
      vs, HH, Wqk, HH, nullptr, QK, HH, HH);
  // 2: streaming attention per (b,a)
  attn_kernel<<<dim3(NPAIR), 256, ATTN_LDS, stream>>>(
      vs, ve, ve_dead, QK, SV, VM, VC);
  // 3: Va = SV @ Wv -> VM[:,128:256]         [2048,128,128]
  gemm_wmma_kernel<0, 0, 0><<<dim3(NPAIR / 16), 256, 0, stream>>>(
      SV, HH, Wv, HH, nullptr, VM + HH, 2 * HH, HH);
  // 4: v_M_final = relu(VM @ W1 + b1) -> out[2048*128:]   [2048,256,128]
  gemm_wmma_kernel<0, 1, 1><<<dim3(NPAIR / 16), 256, 0, stream>>>(
      VM, 2 * HH, W1, HH, b1, out + (size_t)NPAIR * HH, HH, 2 * HH);
  // 5: v_C_final = relu(VC @ W2 + b2) -> out[:2048*128]   [2048,1664,128]
  gemm_wmma_kernel<0, 1, 1><<<dim3(NPAIR / 16), 256, 0, stream>>>(
      VC, (KF + 1) * HH, W2, HH, b2, out, HH, (KF + 1) * HH);
}